// PrototypeLayer_6270652252648
// MI455X (gfx1250) — compile-verified
//
#include <hip/hip_runtime.h>

typedef __attribute__((ext_vector_type(2))) float v2f;
typedef __attribute__((ext_vector_type(8))) float v8f;

namespace {
constexpr int kB = 64;     // batch
constexpr int kC = 2048;   // channels (K dim)
constexpr int kS = 256;    // H*W (N dim)
constexpr int kP = 200;    // prototypes (M dim)
constexpr int kPT = 7;     // ceil(P/32): 32 prototype rows per block
}

// ---------------------------------------------------------------------------
// x_sq[b, s] = sum_c x[b, c, s]^2   (coalesced: lanes walk s, loop walks c)
// ---------------------------------------------------------------------------
__global__ __launch_bounds__(256) void xsq_kernel(const float* __restrict__ x,
                                                  float* __restrict__ xsq) {
    int idx = blockIdx.x * 256 + threadIdx.x;   // idx = b*kS + s,  grid = kB
    int b = idx >> 8;
    int s = idx & 255;
    const float* px = x + (size_t)b * kC * kS + s;
    float acc = 0.f;
#pragma unroll 8
    for (int c = 0; c < kC; ++c) {
        float v = px[(size_t)c * kS];
        acc = fmaf(v, v, acc);
    }
    xsq[idx] = acc;
}

// ---------------------------------------------------------------------------
// p_sq[p] = sum_c proto[p, c]^2   (224 entries, rows >=200 clamped)
// ---------------------------------------------------------------------------
__global__ __launch_bounds__(256) void psq_kernel(const float* __restrict__ proto,
                                                  float* __restrict__ psq) {
    int p = threadIdx.x;
    if (p >= 224) return;
    int pr = p < kP ? p : kP - 1;
    const float* pp = proto + (size_t)pr * kC;
    float acc = 0.f;
#pragma unroll 8
    for (int c = 0; c < kC; ++c) acc = fmaf(pp[c], pp[c], acc);
    psq[p] = acc;
}

// ---------------------------------------------------------------------------
// Main WMMA kernel.
//   grid = (7 tiles of 32 prototype rows, 64 batches), block = 128 (4 waves).
//   Wave w computes the 32x64 stripe: prototypes [pt*32, pt*32+32) x spatial
//   columns [w*64, w*64+64) as 2 (M) x 4 (N) independent 16x16 accumulators.
//   Both M-tiles share the B fragments: per K-step, 2 b64 (A) + 8 b32 (B)
//   vmem ops feed 8 v_wmma_f32_16x16x4_f32 (exact fp32 GEMM on matrix pipe).
//
// Fragment layouts (ISA 7.12.2, 32-bit, wave32):
//   A 16x4 : lanes 0-15 -> M=lane, VGPR{0,1}=K{0,1}; lanes 16-31 -> K{2,3}
//   B 4x16 : lanes 0-15 -> N=lane, VGPR{0,1}=K{0,1}; lanes 16-31 -> K{2,3}
//   C/D    : lane l, vgpr r -> M = r + 8*(l>=16), N = l&15
// ---------------------------------------------------------------------------
__global__ __launch_bounds__(128) void proto_wmma_kernel(
    const float* __restrict__ x, const float* __restrict__ proto,
    const float* __restrict__ xsq, const float* __restrict__ psq,
    float* __restrict__ out_max, float* __restrict__ out_dist,
    float* __restrict__ out_act) {
    __shared__ float red[4][32];   // per-wave row-min partials (32 M rows)

    const int pt   = blockIdx.x;        // prototype 32-row tile, 0..6
    const int b    = blockIdx.y;        // batch
    const int tid  = threadIdx.x;
    const int wave = tid >> 5;          // 0..3
    const int lane = tid & 31;
    const int half = lane >> 4;         // 0 or 1 (selects K pair / M+8)
    const int n    = lane & 15;

    // A fragment rows for the two M tiles (clamped; masked at store time).
    int arow0 = pt * 32 + n;            if (arow0 >= kP) arow0 = kP - 1;
    int arow1 = pt * 32 + 16 + n;       if (arow1 >= kP) arow1 = kP - 1;
    const float* aptr0 = proto + (size_t)arow0 * kC + half * 2;
    const float* aptr1 = proto + (size_t)arow1 * kC + half * 2;

    // B: x[b, k + half*2 + {0,1}, wave*64 + t*16 + n]
    const float* bbase = x + ((size_t)b * kC + half * 2) * kS + wave * 64 + n;

    v8f acc[2][4];
#pragma unroll
    for (int u = 0; u < 2; ++u)
#pragma unroll
        for (int t = 0; t < 4; ++t)
            acc[u][t] = v8f{0.f, 0.f, 0.f, 0.f, 0.f, 0.f, 0.f, 0.f};

#pragma unroll 2
    for (int k = 0; k < kC; k += 4) {
        v2f a0 = *(const v2f*)(aptr0 + k);        // K{k+2h, k+2h+1}, contiguous
        v2f a1 = *(const v2f*)(aptr1 + k);
        const float* bp = bbase + (size_t)k * kS;
#pragma unroll
        for (int t = 0; t < 4; ++t) {
            v2f bv;
            bv.x = bp[t * 16];                    // K = k + 2h
            bv.y = bp[(size_t)kS + t * 16];       // K = k + 2h + 1
            acc[0][t] = __builtin_amdgcn_wmma_f32_16x16x4_f32(
                false, a0, false, bv, (short)0, acc[0][t], false, false);
            acc[1][t] = __builtin_amdgcn_wmma_f32_16x16x4_f32(
                false, a1, false, bv, (short)0, acc[1][t], false, false);
        }
    }

    // Epilogue: d = x_sq + p_sq - 2*cross; write dist & -dist; track row mins.
    float psv[2][8];
    float dmin[2][8];
#pragma unroll
    for (int u = 0; u < 2; ++u)
#pragma unroll
        for (int r = 0; r < 8; ++r) {
            psv[u][r]  = psq[pt * 32 + u * 16 + r + 8 * half];
            dmin[u][r] = 3.4e38f;
        }

#pragma unroll
    for (int t = 0; t < 4; ++t) {
        const int scol = wave * 64 + t * 16 + n;
        const float xsv = xsq[b * kS + scol];
#pragma unroll
        for (int u = 0; u < 2; ++u)
#pragma unroll
            for (int r = 0; r < 8; ++r) {
                float d = fmaf(-2.f, acc[u][t][r], xsv + psv[u][r]);
                dmin[u][r] = fminf(dmin[u][r], d);
                int p = pt * 32 + u * 16 + r + 8 * half;
                if (p < kP) {
                    size_t idx = ((size_t)(b * kP + p)) * kS + scol;
                    out_dist[idx] = d;
                    out_act[idx]  = -d;
                }
            }
    }

    // Reduce across the 16 lanes of each half (rows 0-7 of a tile live in
    // lanes 0-15, rows 8-15 in lanes 16-31; xor masks 1..8 stay in a half).
#pragma unroll
    for (int m = 1; m <= 8; m <<= 1)
#pragma unroll
        for (int u = 0; u < 2; ++u)
#pragma unroll
            for (int r = 0; r < 8; ++r)
                dmin[u][r] = fminf(dmin[u][r], __shfl_xor(dmin[u][r], m, 32));

    if (n == 0) {
#pragma unroll
        for (int u = 0; u < 2; ++u)
#pragma unroll
            for (int r = 0; r < 8; ++r)
                red[wave][u * 16 + r + 8 * half] = dmin[u][r];
    }
    __syncthreads();

    if (tid < 32) {
        float m = fminf(fminf(red[0][tid], red[1][tid]),
                        fminf(red[2][tid], red[3][tid]));
        int p = pt * 32 + tid;
        if (p < kP) out_max[b * kP + p] = -m;   // max similarity = -min distance
    }
}

// ---------------------------------------------------------------------------
extern "C" void kernel_launch(void* const* d_in, const int* in_sizes, int n_in,
                              void* d_out, int out_size, void* d_ws, size_t ws_size,
                              hipStream_t stream) {
    const float* x     = (const float*)d_in[0];   // (B, C, H, W) = (64,2048,16,16)
    const float* proto = (const float*)d_in[1];   // (P, C) = (200, 2048)

    float* xsq = (float*)d_ws;            // kB*kS floats
    float* psq = xsq + kB * kS;           // 224 floats

    float* out_max  = (float*)d_out;                          // (B, P)
    float* out_dist = out_max + kB * kP;                      // (B, P, S)
    float* out_act  = out_dist + (size_t)kB * kP * kS;        // (B, P, H, W)

    xsq_kernel<<<kB, 256, 0, stream>>>(x, xsq);
    psq_kernel<<<1, 256, 0, stream>>>(proto, psq);

    dim3 grid(kPT, kB);
    proto_wmma_kernel<<<grid, 128, 0, stream>>>(x, proto, xsq, psq,
                                                out_max, out_dist, out_act);
}